// PointGNN_55387898249917
// MI455X (gfx1250) — compile-verified
//
#include <hip/hip_runtime.h>
#include <hip/hip_bf16.h>
#include <stdint.h>

// ---------------------------------------------------------------------------
// Point-GNN on gfx1250 (MI455X), fp32 WMMA (V_WMMA_F32_16X16X4_F32) path.
//
// Edge GEMM  concat([dpos, s[src]]) @ Wf  is decomposed as
//   P = s @ Wf[3:,:]                     (vertex GEMM with WMMA)
//   e = relu(P[src] + dpos @ Wf[0:3,:] + bf)   (per-edge rank-3 term)
// followed by an L2-resident atomic-max scatter (relu>=0 => int-max valid).
//
// GEMM kernel: block = 128 threads (4 waves), 64x32 output tile per block,
// 16x32 per wave (two 16x16 WMMA accumulators). A tile is moved to LDS with
// GLOBAL_LOAD_ASYNC_TO_LDS_B32 (ASYNCcnt path, no VGPR round-trip) when the
// builtin exists, else with a batched load-clause/store-clause. Staging is
// branch-free: A uses clamped addresses (garbage is provably harmless: row
// tails are never stored, K tails multiply zero-filled W rows); W is
// zero-padded exactly with clamp+select.
// ---------------------------------------------------------------------------

typedef __attribute__((ext_vector_type(2))) float v2f;
typedef __attribute__((ext_vector_type(8))) float v8f;

#define GEMM_BLOCK 128   // 4 waves of 32

#if __has_builtin(__builtin_amdgcn_global_load_async_to_lds_b32)
#define HAVE_ASYNC_LDS 1
// Builtin signature takes plain int* pointers (per hipcc diagnostic).
#define TO_GLOBAL(p) ((int*)(void*)(p))
#define TO_LDS(p)    ((int*)(void*)(p))
#else
#define HAVE_ASYNC_LDS 0
#endif

__device__ __forceinline__ void wait_async_lds()
{
#if HAVE_ASYNC_LDS
#if __has_builtin(__builtin_amdgcn_s_wait_asynccnt)
    __builtin_amdgcn_s_wait_asynccnt(0);
#else
    asm volatile("s_wait_asynccnt 0x0" ::: "memory");
#endif
#endif
}

template<int KSTEP>
__global__ __launch_bounds__(GEMM_BLOCK)
void wmma_gemm_kernel(const float* __restrict__ A, int lda,
                      const float* __restrict__ W, int ldw,
                      const float* __restrict__ bias,
                      const float* __restrict__ residual, int ldr,
                      float* __restrict__ C, int ldc,
                      int rows, int K, int cols, int do_relu)
{
    constexpr int APAD = KSTEP + 1;          // de-conflict LDS banks on A reads
    constexpr int AE   = 64 * KSTEP / GEMM_BLOCK;   // A elems per thread
    constexpr int WE   = KSTEP * 32 / GEMM_BLOCK;   // W elems per thread
    __shared__ float Alds[64 * APAD];        // 64 rows x KSTEP
    __shared__ float Wlds[KSTEP * 32];       // KSTEP x 32 cols

    const int tid  = threadIdx.x;
    const int lane = tid & 31;
    const int wave = tid >> 5;
    const int half = lane >> 4;              // 0: lanes 0-15, 1: lanes 16-31
    const int l16  = lane & 15;

    const int rowBlock0 = blockIdx.x * 64;
    const int row0      = rowBlock0 + wave * 16;
    const int n0        = blockIdx.y * 32;

    v8f acc0 = {0.f,0.f,0.f,0.f,0.f,0.f,0.f,0.f};
    v8f acc1 = {0.f,0.f,0.f,0.f,0.f,0.f,0.f,0.f};

    for (int k0 = 0; k0 < K; k0 += KSTEP) {
        // ---- A tile: branch-free clamped staging (garbage in pads is harmless)
#if HAVE_ASYNC_LDS
        #pragma unroll
        for (int j = 0; j < AE; ++j) {
            const int i  = tid + j * GEMM_BLOCK;
            const int r  = i / KSTEP;
            const int ck = i % KSTEP;
            const int gr = min(rowBlock0 + r, rows - 1);
            const int gk = min(k0 + ck,      K    - 1);
            __builtin_amdgcn_global_load_async_to_lds_b32(
                TO_GLOBAL(A + (size_t)gr * lda + gk),
                TO_LDS(&Alds[r * APAD + ck]), 0, 0);
        }
#else
        {
            float av[AE];
            #pragma unroll
            for (int j = 0; j < AE; ++j) {
                const int i  = tid + j * GEMM_BLOCK;
                const int gr = min(rowBlock0 + i / KSTEP, rows - 1);
                const int gk = min(k0 + i % KSTEP,        K    - 1);
                av[j] = A[(size_t)gr * lda + gk];
            }
            #pragma unroll
            for (int j = 0; j < AE; ++j) {
                const int i = tid + j * GEMM_BLOCK;
                Alds[(i / KSTEP) * APAD + (i % KSTEP)] = av[j];
            }
        }
#endif
        // ---- W tile: exact zero padding, branchless clamp+select,
        //      load clause then store clause (overlaps the async A copies)
        {
            float wv[WE];
            #pragma unroll
            for (int j = 0; j < WE; ++j) {
                const int i  = tid + j * GEMM_BLOCK;
                const int kk = i >> 5, nn = i & 31;
                const int gk = k0 + kk, gn = n0 + nn;
                const bool ok = (gk < K) && (gn < cols);
                const float v = W[(size_t)min(gk, K - 1) * ldw + min(gn, cols - 1)];
                wv[j] = ok ? v : 0.0f;
            }
            #pragma unroll
            for (int j = 0; j < WE; ++j) {
                const int i = tid + j * GEMM_BLOCK;
                Wlds[(i >> 5) * 32 + (i & 31)] = wv[j];
            }
        }
        wait_async_lds();
        __syncthreads();

        // ---- branch-free WMMA microloop, fully unrolled
        const float* Ar = &Alds[(wave * 16 + l16) * APAD];
        #pragma unroll
        for (int kk = 0; kk < KSTEP; kk += 4) {
            const int kh = kk + half * 2;    // ISA 16x4 f32 A/B lane-half K split
            v2f a, b0, b1;
            a.x  = Ar[kh];
            a.y  = Ar[kh + 1];
            b0.x = Wlds[kh * 32 + l16];
            b0.y = Wlds[(kh + 1) * 32 + l16];
            b1.x = Wlds[kh * 32 + 16 + l16];
            b1.y = Wlds[(kh + 1) * 32 + 16 + l16];
            acc0 = __builtin_amdgcn_wmma_f32_16x16x4_f32(
                       false, a, false, b0, (short)0, acc0, false, false);
            acc1 = __builtin_amdgcn_wmma_f32_16x16x4_f32(
                       false, a, false, b1, (short)0, acc1, false, false);
        }
        __syncthreads();
    }

    if (row0 >= rows) return;                // row tiles are 16-aligned: all-or-nothing
    #pragma unroll
    for (int j = 0; j < 2; ++j) {
        const int ncol = n0 + j * 16 + l16;
        if (ncol >= cols) continue;
        const float bv = bias ? bias[ncol] : 0.0f;
        const v8f acc = j ? acc1 : acc0;
        #pragma unroll
        for (int m8 = 0; m8 < 8; ++m8) {
            // C/D layout: VGPR m8 holds row m8 (lanes 0-15) / m8+8 (lanes 16-31)
            const int row = row0 + m8 + half * 8;
            float v = acc[m8] + bv;
            if (residual) v += residual[(size_t)row * ldr + ncol];
            if (do_relu)  v = fmaxf(v, 0.0f);
            C[(size_t)row * ldc + ncol] = v;
        }
    }
}

// Ragged per-vertex max over key-point features (honors the lookup table).
__global__ void segmax_kp_kernel(const float* __restrict__ feats,
                                 const int* __restrict__ lookup,
                                 float* __restrict__ kpmax,
                                 int n, int m, int sd)
{
    const size_t idx = (size_t)blockIdx.x * blockDim.x + threadIdx.x;
    if (idx >= (size_t)n * sd) return;
    const int v = (int)(idx / sd);
    const int c = (int)(idx % sd);
    const int start = lookup[v];
    const int end   = (v + 1 < n) ? lookup[v + 1] : m;
    float mx = -3.402823466e38f;
    for (int kp = start; kp < end; ++kp)
        mx = fmaxf(mx, feats[(size_t)kp * sd + c]);
    kpmax[idx] = mx;
}

// dx = s @ Wh + bh   ([N,3] tiny GEMV)
__global__ void dx_kernel(const float* __restrict__ s,
                          const float* __restrict__ Wh,   // [sd,3]
                          const float* __restrict__ bh,   // [3]
                          float* __restrict__ dx, int n, int sd)
{
    const int v = blockIdx.x * blockDim.x + threadIdx.x;
    if (v >= n) return;
    float o0 = bh[0], o1 = bh[1], o2 = bh[2];
    const float* sv = s + (size_t)v * sd;
    for (int d = 0; d < sd; ++d) {
        const float x = sv[d];
        o0 = fmaf(x, Wh[d * 3 + 0], o0);
        o1 = fmaf(x, Wh[d * 3 + 1], o1);
        o2 = fmaf(x, Wh[d * 3 + 2], o2);
    }
    dx[v * 3 + 0] = o0; dx[v * 3 + 1] = o1; dx[v * 3 + 2] = o2;
}

__global__ void zero_kernel(float* __restrict__ p, size_t n)
{
    const size_t i = (size_t)blockIdx.x * blockDim.x + threadIdx.x;
    if (i < n) p[i] = 0.0f;
}

// Per-edge message + scatter-max:
//   e_c = relu(P[src][c] + m.Wtop[:,c] + bf[c]);  aggr[dst][c] max= e_c
__global__ void edge_scatter_max_kernel(const int* __restrict__ esrc,
                                        const int* __restrict__ edst,
                                        const float* __restrict__ pos,
                                        const float* __restrict__ dx,
                                        const float* __restrict__ P,
                                        const float* __restrict__ Wtop, // [3,sd]
                                        const float* __restrict__ bf,   // [sd]
                                        float* __restrict__ aggr,
                                        int nedges, int sd)
{
    const int e = blockIdx.x;
    const int c = blockIdx.y * blockDim.x + threadIdx.x;
    if (e >= nedges || c >= sd) return;
    const int s = esrc[e];
    const int d = edst[e];
    const float m0 = pos[s * 3 + 0] - pos[d * 3 + 0] + dx[d * 3 + 0];
    const float m1 = pos[s * 3 + 1] - pos[d * 3 + 1] + dx[d * 3 + 1];
    const float m2 = pos[s * 3 + 2] - pos[d * 3 + 2] + dx[d * 3 + 2];
    float v = P[(size_t)s * sd + c] + bf[c];
    v = fmaf(m0, Wtop[c],          v);
    v = fmaf(m1, Wtop[sd + c],     v);
    v = fmaf(m2, Wtop[2 * sd + c], v);
    v = fmaxf(v, 0.0f);
    // non-negative floats order like their int bit patterns; aggr init = 0
    atomicMax((int*)&aggr[(size_t)d * sd + c], __float_as_int(v));
}

// ---------------------------------------------------------------------------

static inline void launch_gemm(hipStream_t st,
                               const float* A, int lda,
                               const float* W, int ldw,
                               const float* bias,
                               const float* res, int ldr,
                               float* C, int ldc,
                               int rows, int K, int cols, int relu)
{
    dim3 grid((rows + 63) / 64, (cols + 31) / 32);
    if (K <= 4)
        wmma_gemm_kernel<4><<<grid, GEMM_BLOCK, 0, st>>>(
            A, lda, W, ldw, bias, res, ldr, C, ldc, rows, K, cols, relu);
    else
        wmma_gemm_kernel<32><<<grid, GEMM_BLOCK, 0, st>>>(
            A, lda, W, ldw, bias, res, ldr, C, ldc, rows, K, cols, relu);
}

extern "C" void kernel_launch(void* const* d_in, const int* in_sizes, int n_in,
                              void* d_out, int out_size, void* d_ws, size_t ws_size,
                              hipStream_t stream)
{
    // ---- inputs (setup_inputs dict order) ----
    const float* key_points = (const float*)d_in[0];   // [M,4]
    const float* pos        = (const float*)d_in[1];   // [N,3]
    const int*   lookup     = (const int*)  d_in[2];   // [N]
    const int*   edge_index = (const int*)  d_in[3];   // [2,E]
    const float* Wi0 = (const float*)d_in[4];  const float* bi0 = (const float*)d_in[5];
    const float* Wi1 = (const float*)d_in[6];  const float* bi1 = (const float*)d_in[7];
    const float* Wi2 = (const float*)d_in[8];  const float* bi2 = (const float*)d_in[9];
    const float* Wa  = (const float*)d_in[10]; const float* ba  = (const float*)d_in[11];
    const float* Wh  = (const float*)d_in[12]; const float* bh  = (const float*)d_in[13];
    const float* Wf  = (const float*)d_in[14]; const float* bf  = (const float*)d_in[15];
    const float* Wg  = (const float*)d_in[16]; const float* bg  = (const float*)d_in[17];
    const float* Wc1 = (const float*)d_in[18]; const float* bc1 = (const float*)d_in[19];
    const float* Wc2 = (const float*)d_in[20]; const float* bc2 = (const float*)d_in[21];
    const float* Wl1 = (const float*)d_in[22]; const float* bl1 = (const float*)d_in[23];
    const float* Wl2 = (const float*)d_in[24]; const float* bl2 = (const float*)d_in[25];
    const float* Wl3 = (const float*)d_in[26]; const float* bl3 = (const float*)d_in[27];

    // ---- problem sizes (derived from input shapes) ----
    const int M  = in_sizes[0] / 4;        // key points
    const int N  = in_sizes[1] / 3;        // vertices
    const int E  = in_sizes[3] / 2;        // edges
    const int SD = in_sizes[11];           // state dim (300)
    const int T  = in_sizes[13] / 3;       // GNN iterations
    const int NC = in_sizes[21];           // classes
    const int H1 = in_sizes[19];           // cls hidden (64)
    const int L1 = in_sizes[23] / NC;      // loc hidden 1 (64)
    const int L2 = in_sizes[25] / NC;      // loc hidden 2 (64)
    const int L3 = in_sizes[27] / NC;      // loc out (7)

    // ---- workspace partition (floats) ----
    float* ws = (float*)d_ws;
    float* f_h0    = ws;                       // M * 64
    float* f_h1    = f_h0   + (size_t)M * 64;  // M * 128
    float* f_feat  = f_h1   + (size_t)M * 128; // M * SD
    float* f_kpmax = f_feat + (size_t)M * SD;  // N * SD
    float* f_s     = f_kpmax+ (size_t)N * SD;  // N * SD
    float* f_P     = f_s    + (size_t)N * SD;  // N * SD
    float* f_aggr  = f_P    + (size_t)N * SD;  // N * SD
    float* f_dx    = f_aggr + (size_t)N * SD;  // N * 3
    float* f_c1    = f_dx   + (size_t)N * 3;   // N * H1
    float* f_l1    = f_c1   + (size_t)N * H1;  // N * L1
    float* f_l2    = f_l1   + (size_t)N * L1;  // N * L2
    (void)ws_size; (void)n_in; (void)out_size;

    // ---- initial key-point MLP: 4 -> 64 -> 128 -> SD, relu ----
    launch_gemm(stream, key_points, 4,  Wi0, 64,  bi0, nullptr, 0, f_h0, 64,  M, 4,   64, 1);
    launch_gemm(stream, f_h0,       64, Wi1, 128, bi1, nullptr, 0, f_h1, 128, M, 64, 128, 1);
    launch_gemm(stream, f_h1,      128, Wi2, SD,  bi2, nullptr, 0, f_feat, SD, M, 128, SD, 1);

    // ---- ragged segment-max over key points, then s = relu(kpmax @ Wa + ba) ----
    {
        const size_t tot = (size_t)N * SD;
        segmax_kp_kernel<<<dim3((unsigned)((tot + 255) / 256)), 256, 0, stream>>>(
            f_feat, lookup, f_kpmax, N, M, SD);
    }
    launch_gemm(stream, f_kpmax, SD, Wa, SD, ba, nullptr, 0, f_s, SD, N, SD, SD, 1);

    // ---- GNN iterations ----
    const size_t WfStride = (size_t)(SD + 3) * SD;
    for (int t = 0; t < T; ++t) {
        const float* Wf_t   = Wf + (size_t)t * WfStride;   // [SD+3, SD]
        const float* Wf_top = Wf_t;                        // rows 0..2
        const float* Wf_bot = Wf_t + 3 * SD;               // rows 3..SD+2

        // dx = s @ Wh[t] + bh[t]
        dx_kernel<<<dim3((N + 255) / 256), 256, 0, stream>>>(
            f_s, Wh + (size_t)t * SD * 3, bh + t * 3, f_dx, N, SD);

        // P = s @ Wf_bot   (bias folded into edge kernel)
        launch_gemm(stream, f_s, SD, Wf_bot, SD, nullptr, nullptr, 0, f_P, SD, N, SD, SD, 0);

        // aggr = 0 ; scatter-max edge messages into aggr
        {
            const size_t tot = (size_t)N * SD;
            zero_kernel<<<dim3((unsigned)((tot + 255) / 256)), 256, 0, stream>>>(f_aggr, tot);
        }
        edge_scatter_max_kernel<<<dim3(E, (SD + 127) / 128), 128, 0, stream>>>(
            edge_index, edge_index + E, pos, f_dx, f_P,
            Wf_top, bf + (size_t)t * SD, f_aggr, E, SD);

        // s = s + aggr @ Wg[t] + bg[t]   (residual, no relu; in-place safe:
        // each output element is read-modify-written by exactly one lane)
        launch_gemm(stream, f_aggr, SD, Wg + (size_t)t * SD * SD, SD,
                    bg + (size_t)t * SD, f_s, SD, f_s, SD, N, SD, SD, 0);
    }

    // ---- heads ----
    float* cls_out = (float*)d_out;                 // [N, NC]
    float* reg_out = cls_out + (size_t)N * NC;      // [N, NC*L3]

    launch_gemm(stream, f_s,  SD, Wc1, H1, bc1, nullptr, 0, f_c1, H1, N, SD, H1, 1);
    launch_gemm(stream, f_c1, H1, Wc2, NC, bc2, nullptr, 0, cls_out, NC, N, H1, NC, 1);

    for (int c = 0; c < NC; ++c) {
        launch_gemm(stream, f_s,  SD, Wl1 + (size_t)c * SD * L1, L1,
                    bl1 + (size_t)c * L1, nullptr, 0, f_l1, L1, N, SD, L1, 1);
        launch_gemm(stream, f_l1, L1, Wl2 + (size_t)c * L1 * L2, L2,
                    bl2 + (size_t)c * L2, nullptr, 0, f_l2, L2, N, L1, L2, 1);
        // stripe class c's 7 outputs into reg_pred[:, c*7 : c*7+7]
        launch_gemm(stream, f_l2, L2, Wl3 + (size_t)c * L2 * L3, L3,
                    bl3 + (size_t)c * L3, nullptr, 0, reg_out + c * L3, NC * L3,
                    N, L2, L3, 1);
    }
}